// BLOutputLayer_74594991997075
// MI455X (gfx1250) — compile-verified
//
#include <hip/hip_runtime.h>

// BLOutputLayer forward: rule-book gather
//   out[row, :] = features[rules[row], :],  features: [N_ACTIVE, 64] f32
//
// Memory-bound kernel. Strategy:
//  - 16 lanes per row, one float4 per lane -> 256B contiguous burst per row
//    (best possible coalescing for a row gather), 2 rows per wave32.
//  - features reads: default (RT) temporal hint -> 48.8 MiB table stays
//    resident in the 192 MB L2 across the 2.6x average reuse.
//  - output stores: non-temporal (TH=NT on global_store_b128) -> the 128 MiB
//    write-once stream does not evict the features table from L2.
//  - rules reads: non-temporal loads (read exactly once).

typedef __attribute__((ext_vector_type(4))) float f32x4;

constexpr int C          = 64;       // channels per row
constexpr int VEC        = 4;        // floats per vector op
constexpr int LANES_ROW  = C / VEC;  // 16 lanes cover one row
constexpr int BLOCK      = 256;      // 8 waves per block (wave32)

__global__ __launch_bounds__(BLOCK) void
bl_output_gather_kernel(const float* __restrict__ features,
                        const int*   __restrict__ rules,
                        float*       __restrict__ out,
                        int nRows)
{
    const int t     = blockIdx.x * BLOCK + threadIdx.x;
    const int row   = t >> 4;          // 16 threads per row
    const int chunk = t & (LANES_ROW - 1);
    if (row >= nRows) return;

    // All 16 lanes of a half-wave hit the same address; HW merges into one
    // transaction. NT hint: rules are streamed, never re-read.
    const int idx = __builtin_nontemporal_load(rules + row);

    // Gathered read of one contiguous 256B row; keep RT so the feature table
    // (48.8 MiB) remains L2-resident.
    const f32x4* __restrict__ src =
        (const f32x4*)(features + (long long)idx * C);
    f32x4 v = src[chunk];

    // Sequential streaming store with non-temporal hint
    // (emits global_store_b128 th:NT) to avoid polluting L2.
    f32x4* __restrict__ dst = (f32x4*)(out + (long long)row * C);
    __builtin_nontemporal_store(v, dst + chunk);
}

extern "C" void kernel_launch(void* const* d_in, const int* in_sizes, int n_in,
                              void* d_out, int out_size, void* d_ws, size_t ws_size,
                              hipStream_t stream)
{
    const float* features = (const float*)d_in[0];   // [N_ACTIVE, 64] f32
    const int*   rules    = (const int*)d_in[1];     // [N_ROWS] int
    float*       out      = (float*)d_out;           // [N_ROWS, 64] f32

    const int nRows = in_sizes[1];                   // 524288
    const long long totalThreads = (long long)nRows * LANES_ROW;
    const int grid = (int)((totalThreads + BLOCK - 1) / BLOCK);

    bl_output_gather_kernel<<<grid, BLOCK, 0, stream>>>(features, rules, out, nRows);
}